// MaxMinComp_80049600463150
// MI455X (gfx1250) — compile-verified
//
#include <hip/hip_runtime.h>

// Max-min fuzzy composition: out[b,o] = max_i min(x[b,i], W[i,o])
// B=1024, IN=512, OUT=512, all f32.
// VALU-bound (no WMMA path for a (max,min) semiring). CDNA5-specific path:
// async global->LDS tile loads (ASYNCcnt) + s_wait_asynccnt.

#define B_DIM   1024
#define IN_F    512
#define OUT_F   512

#define TILE_M  64
#define TILE_N  64
#define TILE_K  16
#define PAD     4      // row = 68 floats = 272 B, 16B-aligned rows

typedef __attribute__((ext_vector_type(4))) int v4i;

typedef __attribute__((address_space(1))) int  g_i32;   // global int
typedef __attribute__((address_space(3))) int  l_i32;   // LDS int
typedef __attribute__((address_space(1))) v4i  g_v4i;   // global int4
typedef __attribute__((address_space(3))) v4i  l_v4i;   // LDS int4

#if defined(__AMDGCN__) && \
    __has_builtin(__builtin_amdgcn_global_load_async_to_lds_b32) && \
    __has_builtin(__builtin_amdgcn_global_load_async_to_lds_b128)
#define USE_ASYNC 1
#else
#define USE_ASYNC 0
#endif

#if defined(__AMDGCN__) && __has_builtin(__builtin_amdgcn_s_wait_asynccnt)
#define WAIT_ASYNC() __builtin_amdgcn_s_wait_asynccnt(0)
#else
#define WAIT_ASYNC() asm volatile("s_wait_asynccnt 0" ::: "memory")
#endif

__global__ __launch_bounds__(256) void
maxmin_comp_kernel(const float* __restrict__ x,
                   const float* __restrict__ w,
                   float* __restrict__ out) {
  __shared__ float xs[TILE_K][TILE_M + PAD];  // xs[k][row]  (transposed x tile)
  __shared__ float ws[TILE_K][TILE_N + PAD];  // ws[k][col]

  const int tid = threadIdx.x;
  const int tx  = tid & 15;   // 16 threads across N
  const int ty  = tid >> 4;   // 16 threads across M
  const int row0 = blockIdx.x * TILE_M;   // 16 blocks in M
  const int col0 = blockIdx.y * TILE_N;   // 8 blocks in N

  float acc[4][4];
#pragma unroll
  for (int i = 0; i < 4; ++i)
#pragma unroll
    for (int j = 0; j < 4; ++j)
      acc[i][j] = -__builtin_huge_valf();

  for (int k0 = 0; k0 < IN_F; k0 += TILE_K) {
    if (k0) __syncthreads();   // previous compute done before LDS overwrite

#if USE_ASYNC
    // x tile (transposed into xs[k][i]): 1024 elems, 4 async b32 per thread.
    // Lane mapping k-fast -> two contiguous 64B bursts per wave.
#pragma unroll
    for (int p = 0; p < 4; ++p) {
      const int idx = tid + p * 256;
      const int k = idx & 15;
      const int i = idx >> 4;
      __builtin_amdgcn_global_load_async_to_lds_b32(
          (g_i32*)(x + (size_t)(row0 + i) * IN_F + (k0 + k)),
          (l_i32*)&xs[k][i], 0, 0);
    }
    // w tile: row-contiguous in both global and LDS -> one async b128 each.
    {
      const int j4 = (tid & 15) * 4;
      const int k  = tid >> 4;
      __builtin_amdgcn_global_load_async_to_lds_b128(
          (g_v4i*)(w + (size_t)(k0 + k) * OUT_F + (col0 + j4)),
          (l_v4i*)&ws[k][j4], 0, 0);
    }
    WAIT_ASYNC();              // my async loads landed in LDS
#else
#pragma unroll
    for (int p = 0; p < 4; ++p) {
      const int idx = tid + p * 256;
      const int k = idx & 15;
      const int i = idx >> 4;
      xs[k][i] = x[(size_t)(row0 + i) * IN_F + (k0 + k)];
    }
    {
      const int j4 = (tid & 15) * 4;
      const int k  = tid >> 4;
      *(float4*)&ws[k][j4] =
          *(const float4*)&w[(size_t)(k0 + k) * OUT_F + (col0 + j4)];
    }
#endif
    __syncthreads();           // everyone's tile data visible

    // Inner loop: 2 K-steps fused so the accumulate folds to v_max3_f32:
    // acc = max3(acc, min(a0,b0), min(a1,b1)).  48 VALU + 4 ds_load_b128
    // per 2 K-steps per thread (16 outputs).
#pragma unroll
    for (int k = 0; k < TILE_K; k += 2) {
      const float4 a0 = *(const float4*)&xs[k][ty * 4];
      const float4 b0 = *(const float4*)&ws[k][tx * 4];
      const float4 a1 = *(const float4*)&xs[k + 1][ty * 4];
      const float4 b1 = *(const float4*)&ws[k + 1][tx * 4];
      const float A0[4] = {a0.x, a0.y, a0.z, a0.w};
      const float B0[4] = {b0.x, b0.y, b0.z, b0.w};
      const float A1[4] = {a1.x, a1.y, a1.z, a1.w};
      const float B1[4] = {b1.x, b1.y, b1.z, b1.w};
#pragma unroll
      for (int i = 0; i < 4; ++i) {
#pragma unroll
        for (int j = 0; j < 4; ++j) {
          const float m0 = fminf(A0[i], B0[j]);
          const float m1 = fminf(A1[i], B1[j]);
          acc[i][j] = fmaxf(acc[i][j], fmaxf(m0, m1));  // -> v_max3_f32
        }
      }
    }
  }

  // Store 4 rows x float4 per thread, fully coalesced.
#pragma unroll
  for (int i = 0; i < 4; ++i) {
    float4 v;
    v.x = acc[i][0]; v.y = acc[i][1]; v.z = acc[i][2]; v.w = acc[i][3];
    *(float4*)&out[(size_t)(row0 + ty * 4 + i) * OUT_F + (col0 + tx * 4)] = v;
  }
}

extern "C" void kernel_launch(void* const* d_in, const int* in_sizes, int n_in,
                              void* d_out, int out_size, void* d_ws, size_t ws_size,
                              hipStream_t stream) {
  (void)in_sizes; (void)n_in; (void)d_ws; (void)ws_size; (void)out_size;
  const float* x = (const float*)d_in[0];   // [1024, 512]
  const float* w = (const float*)d_in[1];   // [512, 512]
  float* out = (float*)d_out;               // [1024, 512]

  dim3 grid(B_DIM / TILE_M, OUT_F / TILE_N);  // (16, 8)
  dim3 block(256);
  maxmin_comp_kernel<<<grid, block, 0, stream>>>(x, w, out);
}